// L_Grad_2886218022923
// MI455X (gfx1250) — compile-verified
//
#include <hip/hip_runtime.h>
#include <stdint.h>

// ---------------------------------------------------------------------------
// Problem constants (fixed by setup_inputs: [8,3,1024,1024] fp32)
// ---------------------------------------------------------------------------
namespace {
constexpr int IMG_H  = 1024;
constexpr int IMG_W  = 1024;
constexpr int NBATCH = 8;
constexpr int PLANE  = IMG_H * IMG_W;          // 1048576 per-batch channel-0 plane
constexpr int TX = 32, TY = 32;                // output tile per block
constexpr int TILE = TX + 2;                   // 34 (halo of 1)
constexpr int TSZ  = TILE * TILE;              // 1156
constexpr int NPLANES = 2 * NBATCH;            // (image A|B) x batch = 16 select problems
}

// ---------------------------------------------------------------------------
// CDNA5 async global->LDS staging (ASYNCcnt path). Builtin takes global int* /
// LDS int* (per hipcc diagnostics). Guarded so the file always compiles.
// ---------------------------------------------------------------------------
#if defined(__gfx1250__) && __has_builtin(__builtin_amdgcn_global_load_async_to_lds_b32)
#define USE_ASYNC_LDS 1
#else
#define USE_ASYNC_LDS 0
#endif

typedef __attribute__((address_space(1))) int as1_int;
typedef __attribute__((address_space(3))) int as3_int;

__device__ __forceinline__ void cp_g2lds_b32(float* lds_dst, const float* gsrc) {
#if USE_ASYNC_LDS
  // (global src, lds dst, imm offset, imm cpol)
  __builtin_amdgcn_global_load_async_to_lds_b32(
      (as1_int*)gsrc, (as3_int*)lds_dst, 0, 0);
#else
  *lds_dst = *gsrc;
#endif
}

__device__ __forceinline__ void cp_wait_async() {
#if USE_ASYNC_LDS
#if __has_builtin(__builtin_amdgcn_s_wait_asynccnt)
  __builtin_amdgcn_s_wait_asynccnt(0);
#else
  asm volatile("s_wait_asynccnt 0" ::: "memory");
#endif
#endif
}

// ---------------------------------------------------------------------------
// Sobel |Gx|+|Gy| at tile-local position; s points at tile[r*34 + tx].
// KX = [-1 0 1; -2 0 2; -1 0 1], KY = [1 2 1; 0 0 0; -1 -2 -1]
// (conv_general_dilated = cross-correlation, no kernel flip)
// ---------------------------------------------------------------------------
__device__ __forceinline__ float sobel_at(const float* s) {
  float a00 = s[0],       a01 = s[1],  a02 = s[2];
  float a10 = s[TILE],                 a12 = s[TILE + 2];
  float a20 = s[2*TILE],  a21 = s[2*TILE + 1], a22 = s[2*TILE + 2];
  float gx = (a02 - a00) + 2.0f * (a12 - a10) + (a22 - a20);
  float gy = (a00 + 2.0f * a01 + a02) - (a20 + 2.0f * a21 + a22);
  return fabsf(gx) + fabsf(gy);
}

// ---------------------------------------------------------------------------
// Kernel 1: fused Sobel of A/B/F (channel 0) + loss partials + radix PASS 0
// histograms (top byte) for gA/gB while values are still in registers.
// grid = (32, 32, 8), block = 256 (8 waves of 32)
// ---------------------------------------------------------------------------
__global__ __launch_bounds__(256)
void sobel_kernel(const float* __restrict__ imgA, const float* __restrict__ imgB,
                  const float* __restrict__ imgF,
                  float* __restrict__ gA, float* __restrict__ gB,
                  float* __restrict__ partials, unsigned* __restrict__ ghist) {
  __shared__ float s[3 * TSZ];     // A tile | B tile | F tile, each 34x34
  __shared__ float red[256];
  __shared__ unsigned hA[256], hB[256];

  const int tid = threadIdx.x;
  const int b   = blockIdx.z;
  const int gx0 = blockIdx.x * TX - 1;
  const int gy0 = blockIdx.y * TY - 1;
  const size_t ch0 = (size_t)b * 3u * (size_t)PLANE;  // channel 0 of batch b
  const float* srcA = imgA + ch0;
  const float* srcB = imgB + ch0;
  const float* srcF = imgF + ch0;

  hA[tid] = 0u;
  hB[tid] = 0u;

  // Stage 3 haloed tiles into LDS (async on CDNA5; zero-fill the padding).
  for (int i = tid; i < 3 * TSZ; i += 256) {
    int img = i / TSZ;
    int rem = i - img * TSZ;
    int r = rem / TILE;
    int c = rem - r * TILE;
    int gy = gy0 + r, gx = gx0 + c;
    const float* src = (img == 0) ? srcA : ((img == 1) ? srcB : srcF);
    if ((unsigned)gy < (unsigned)IMG_H && (unsigned)gx < (unsigned)IMG_W) {
      cp_g2lds_b32(&s[i], src + gy * IMG_W + gx);
    } else {
      s[i] = 0.0f;
    }
  }
  cp_wait_async();
  __syncthreads();

  const float* sA = s;
  const float* sB = s + TSZ;
  const float* sF = s + 2 * TSZ;

  const int tx = tid & 31;
  const int ty = tid >> 5;          // 0..7; each thread does 4 rows
  float acc = 0.0f;

  for (int j = 0; j < 4; ++j) {
    int r = ty + 8 * j;             // 0..31
    int off = r * TILE + tx;        // window top-left in tile space
    float vA = sobel_at(sA + off);
    float vB = sobel_at(sB + off);
    float vF = sobel_at(sF + off);

    int y = blockIdx.y * TY + r;
    int x = blockIdx.x * TX + tx;
    int o = b * PLANE + y * IMG_W + x;
    gA[o] = vA;
    gB[o] = vB;
    acc += fabsf(vF - fmaxf(vA, vB));

    // Radix pass 0: histogram of the most significant byte (values >= 0, so
    // uint bit order == float order).
    atomicAdd(&hA[__float_as_uint(vA) >> 24], 1u);
    atomicAdd(&hB[__float_as_uint(vB) >> 24], 1u);
  }

  // Deterministic block reduction (fixed tree order).
  red[tid] = acc;
  __syncthreads();
  for (int st = 128; st > 0; st >>= 1) {
    if (tid < st) red[tid] += red[tid + st];
    __syncthreads();
  }
  if (tid == 0) {
    int bid = (blockIdx.z * gridDim.y + blockIdx.y) * gridDim.x + blockIdx.x;
    partials[bid] = red[0];
  }

  // Flush pass-0 histograms: plane index = img*8 + b (img: 0=A, 1=B).
  if (hA[tid]) atomicAdd(&ghist[(0 * NPLANES + b) * 256 + tid], hA[tid]);
  if (hB[tid]) atomicAdd(&ghist[(0 * NPLANES + NBATCH + b) * 256 + tid], hB[tid]);
}

// ---------------------------------------------------------------------------
// Kernel 0: init radix-select state and histograms.
// state[t] = {prefix_bits, k_remaining}; t = img*8 + batch (img: 0=A, 1=B)
// ---------------------------------------------------------------------------
__global__ void init_kernel(unsigned* __restrict__ ghist, uint2* __restrict__ state,
                            const int* __restrict__ kptr) {
  int tid = blockIdx.x * blockDim.x + threadIdx.x;
  int n = 4 * NPLANES * 256;
  for (int i = tid; i < n; i += gridDim.x * blockDim.x) ghist[i] = 0u;
  if (tid < NPLANES) state[tid] = make_uint2(0u, (unsigned)kptr[0]);
}

// ---------------------------------------------------------------------------
// Kernel 2: radix-select histogram pass p in {1,2,3} (byte 3-p, MSB first).
// grid = (256, 1, 16): 256 blocks x 4096 elems per plane; block = 256.
// float4 (B128) loads; gA/gB (67 MB) are expected L2-resident (192 MB L2).
// ---------------------------------------------------------------------------
__global__ __launch_bounds__(256)
void hist_kernel(const float* __restrict__ gA, const float* __restrict__ gB,
                 const uint2* __restrict__ state, unsigned* __restrict__ ghist,
                 int p, int shift) {
  __shared__ unsigned h[256];
  const int tid = threadIdx.x;
  h[tid] = 0u;
  __syncthreads();

  const int plane = blockIdx.z;                 // 0..15
  const float* base = ((plane >> 3) ? gB : gA) + (plane & 7) * PLANE
                      + blockIdx.x * 4096;
  const unsigned pref = state[plane].x;
  const int hs = shift + 8;                     // bits above the current byte
  const unsigned hp = pref >> hs;               // p >= 1 here, so hs <= 24

  const float4* base4 = (const float4*)base;    // 16B-aligned (16 KB block offset)
  for (int k = 0; k < 4; ++k) {
    float4 v = base4[k * 256 + tid];
    unsigned u0 = __float_as_uint(v.x), u1 = __float_as_uint(v.y);
    unsigned u2 = __float_as_uint(v.z), u3 = __float_as_uint(v.w);
    if ((u0 >> hs) == hp) atomicAdd(&h[(u0 >> shift) & 255u], 1u);
    if ((u1 >> hs) == hp) atomicAdd(&h[(u1 >> shift) & 255u], 1u);
    if ((u2 >> hs) == hp) atomicAdd(&h[(u2 >> shift) & 255u], 1u);
    if ((u3 >> hs) == hp) atomicAdd(&h[(u3 >> shift) & 255u], 1u);
  }
  __syncthreads();
  if (h[tid]) atomicAdd(&ghist[(p * NPLANES + plane) * 256 + tid], h[tid]);
}

// ---------------------------------------------------------------------------
// Kernel 3: per-pass scan — pick the byte containing the k-th smallest.
// One wave; thread t handles select problem t.
// ---------------------------------------------------------------------------
__global__ void scan_kernel(uint2* __restrict__ state,
                            const unsigned* __restrict__ ghist,
                            int p, int shift) {
  int t = threadIdx.x;
  if (t >= NPLANES) return;
  uint2 st = state[t];
  unsigned prefix = st.x, kRem = st.y;
  const unsigned* h = &ghist[(p * NPLANES + t) * 256];
  for (int c = 0; c < 256; ++c) {
    unsigned cnt = h[c];
    if (kRem <= cnt) { prefix |= ((unsigned)c) << shift; break; }
    kRem -= cnt;
  }
  state[t] = make_uint2(prefix, kRem);
}

// ---------------------------------------------------------------------------
// Kernel 4: masks. out layout: [loss][maskA flat][maskB flat].
// Each thread handles 4 consecutive elements (float4 load, NT scalar stores —
// the output start is offset by the loss scalar, so 16B stores can't align).
// grid = 16384 x 256.
// ---------------------------------------------------------------------------
__global__ __launch_bounds__(256)
void mask_kernel(const float* __restrict__ gA, const float* __restrict__ gB,
                 const uint2* __restrict__ state, float* __restrict__ out) {
  unsigned t  = blockIdx.x * 256u + threadIdx.x;      // < 4194304
  unsigned e0 = t * 4u;                               // element index, 16B aligned
  int img = (int)(e0 >> 23);
  int b   = (int)((e0 >> 20) & 7u);
  int i   = (int)(e0 & (PLANE - 1));
  const float* g = (img ? gB : gA) + b * PLANE;
  float4 v = *(const float4*)(g + i);
  unsigned thr = state[img * NBATCH + b].x;           // exact k-th value bits
  float* o = out + 1 + e0;
  __builtin_nontemporal_store((__float_as_uint(v.x) >= thr) ? 1.0f : 0.0f, o + 0);
  __builtin_nontemporal_store((__float_as_uint(v.y) >= thr) ? 1.0f : 0.0f, o + 1);
  __builtin_nontemporal_store((__float_as_uint(v.z) >= thr) ? 1.0f : 0.0f, o + 2);
  __builtin_nontemporal_store((__float_as_uint(v.w) >= thr) ? 1.0f : 0.0f, o + 3);
}

// ---------------------------------------------------------------------------
// Kernel 5: final deterministic loss reduction of 8192 partials.
// ---------------------------------------------------------------------------
__global__ __launch_bounds__(256)
void loss_kernel(const float* __restrict__ partials, float* __restrict__ out) {
  __shared__ float red[256];
  int tid = threadIdx.x;
  float s = 0.0f;
  for (int i = tid; i < 8192; i += 256) s += partials[i];  // fixed order
  red[tid] = s;
  __syncthreads();
  for (int st = 128; st > 0; st >>= 1) {
    if (tid < st) red[tid] += red[tid + st];
    __syncthreads();
  }
  if (tid == 0) out[0] = red[0] / 8388608.0f;   // mean over [8,1,1024,1024]
}

// ---------------------------------------------------------------------------
// Host entry. Workspace layout:
//   gA: 8x1M floats | gB: 8x1M floats | 8192 float partials |
//   4*16*256 uint histograms | 16 uint2 state.  Total ~67.2 MB.
// ---------------------------------------------------------------------------
extern "C" void kernel_launch(void* const* d_in, const int* in_sizes, int n_in,
                              void* d_out, int out_size, void* d_ws, size_t ws_size,
                              hipStream_t stream) {
  (void)in_sizes; (void)n_in; (void)out_size; (void)ws_size;
  const float* A = (const float*)d_in[0];
  const float* B = (const float*)d_in[1];
  const float* F = (const float*)d_in[2];
  const int* kptr = (const int*)d_in[3];
  float* out = (float*)d_out;

  float* gA = (float*)d_ws;
  float* gB = gA + (size_t)NBATCH * PLANE;
  float* partials = gB + (size_t)NBATCH * PLANE;         // 8192 entries
  unsigned* ghist = (unsigned*)(partials + 8192);        // 4*16*256 entries
  uint2* state = (uint2*)(ghist + 4 * NPLANES * 256);    // 16 entries

  init_kernel<<<64, 256, 0, stream>>>(ghist, state, kptr);

  dim3 grid(IMG_W / TX, IMG_H / TY, NBATCH);             // 32 x 32 x 8
  sobel_kernel<<<grid, 256, 0, stream>>>(A, B, F, gA, gB, partials, ghist);
  scan_kernel<<<1, 32, 0, stream>>>(state, ghist, 0, 24); // consume fused pass 0

  for (int p = 1; p < 4; ++p) {
    int shift = 24 - 8 * p;
    hist_kernel<<<dim3(256, 1, NPLANES), 256, 0, stream>>>(gA, gB, state, ghist, p, shift);
    scan_kernel<<<1, 32, 0, stream>>>(state, ghist, p, shift);
  }

  mask_kernel<<<16384, 256, 0, stream>>>(gA, gB, state, out);
  loss_kernel<<<1, 256, 0, stream>>>(partials, out);
}